// MyNet_3994319585649
// MI455X (gfx1250) — compile-verified
//
#include <hip/hip_runtime.h>
#include <math.h>

// ---------------------------------------------------------------------------
// Problem constants (from reference): features (128 x 2048) f32, TEMP=0.07
// Algebra: with f row-l2-normalized, define G_pq[i,j] = sum_k f[i,k]^p f[j,k]^q.
//   d_self[i,j]  = (G11 + G41) / (TEMP * max(sqrt(1 + 2*G32 + G62), eps))
//   d_other[i,j] = (G11 + G32) / (TEMP * max(sqrt(1 + 2*G23 + G44), eps))
//   vs[i] = sum_j exp(d_self) + 3*exp(d_other)
//   loss  = (1/B) sum_i log(vs[i]) - (1/B^2) sum_ij d_self[i,j]
// ---------------------------------------------------------------------------

#define BDIM 128
#define KDIM 2048
#define TEMP_F 0.07f

typedef float v2f __attribute__((ext_vector_type(2)));
typedef float v8f __attribute__((ext_vector_type(8)));

static __device__ __forceinline__ v8f wmma_f32(v2f a, v2f b, v8f c) {
  // V_WMMA_F32_16X16X4_F32: D = A(16x4) * B(4x16) + C(16x16), full f32
  return __builtin_amdgcn_wmma_f32_16x16x4_f32(
      /*neg_a=*/false, a, /*neg_b=*/false, b,
      /*c_mod=*/(short)0, c, /*reuse_a=*/false, /*reuse_b=*/false);
}

// One k-step (K+=4): scale to normalized f, build powers, 6 WMMA accumulations
static __device__ __forceinline__ void gram_step(v2f a, v2f b, float sa,
                                                 float sb, v8f (&acc)[6]) {
  a = a * sa;
  b = b * sb;
  v2f a2 = a * a, a3 = a2 * a, a4 = a2 * a2, a6 = a3 * a3;
  v2f b2 = b * b, b3 = b2 * b, b4 = b2 * b2;
  acc[0] = wmma_f32(a,  b,  acc[0]);  // G11
  acc[1] = wmma_f32(a4, b,  acc[1]);  // G41
  acc[2] = wmma_f32(a3, b2, acc[2]);  // G32
  acc[3] = wmma_f32(a6, b2, acc[3]);  // G62
  acc[4] = wmma_f32(a4, b4, acc[4]);  // G44
  acc[5] = wmma_f32(a2, b3, acc[5]);  // G23
}

// Load one 16-k chunk (4 k-steps) of A and B fragments for this lane
static __device__ __forceinline__ void load_chunk(
    const float* __restrict__ Ap, const float* __restrict__ Bp, int k,
    int koff, v2f (&a)[4], v2f (&b)[4]) {
#pragma unroll
  for (int t = 0; t < 4; ++t) {
    a[t] = *(const v2f*)(Ap + k + 4 * t + koff);
    b[t] = *(const v2f*)(Bp + k + 4 * t + koff);
  }
}

static __device__ __forceinline__ void compute_chunk(const v2f (&a)[4],
                                                     const v2f (&b)[4],
                                                     float sa, float sb,
                                                     v8f (&acc)[6]) {
#pragma unroll
  for (int t = 0; t < 4; ++t) gram_step(a[t], b[t], sa, sb, acc);
}

// ---------------------------------------------------------------------------
// Kernel 1: per-row inverse L2 norm: invn[row] = 1 / max(||x_row||, 1e-12)
// ---------------------------------------------------------------------------
__global__ __launch_bounds__(256) void row_invnorm_kernel(
    const float* __restrict__ F, float* __restrict__ invn) {
  const int row = blockIdx.x;
  const float* __restrict__ p = F + row * KDIM;
  float s = 0.0f;
  for (int k = threadIdx.x; k < KDIM; k += 256) {
    float v = p[k];
    s += v * v;
  }
  __shared__ float red[256];
  red[threadIdx.x] = s;
  __syncthreads();
  for (int st = 128; st > 0; st >>= 1) {
    if ((int)threadIdx.x < st) red[threadIdx.x] += red[threadIdx.x + st];
    __syncthreads();
  }
  if (threadIdx.x == 0) invn[row] = 1.0f / fmaxf(sqrtf(red[0]), 1e-12f);
}

// ---------------------------------------------------------------------------
// Kernel 2: one 16x16 output tile per block; 4 waves split K (512 each).
// Software-pipelined (double-buffered) K loop: a full chunk of 24 WMMAs +
// 36 pk_muls sits between each load clause and its consuming wait, hiding L2
// latency even at 1 wave/SIMD occupancy.
//
// f32 16x4 A-fragment layout (ISA 7.12.2): lane&15 = M, v0 = K=(lane<16?0:2),
// v1 = K+1.  B 4x16 mirrors it with lane&15 = N.  Both map to a contiguous
// float2 load from the owning row of F.
// ---------------------------------------------------------------------------
__global__ __launch_bounds__(128) void gram_tile_kernel(
    const float* __restrict__ F, const float* __restrict__ invn,
    float* __restrict__ dself, float* __restrict__ dother) {
  const int lane = threadIdx.x & 31;
  const int wave = threadIdx.x >> 5;          // 0..3
  const int bi = blockIdx.y;                  // row tile
  const int bj = blockIdx.x;                  // col tile
  const int sub = lane & 15;
  const int koff = (lane >> 4) << 1;          // 0 or 2

  const int rowA = bi * 16 + sub;             // M index for A fragment
  const int rowB = bj * 16 + sub;             // N index for B fragment
  const float sa = invn[rowA];
  const float sb = invn[rowB];
  const float* __restrict__ Ap = F + rowA * KDIM;
  const float* __restrict__ Bp = F + rowB * KDIM;

  v8f acc[6];
#pragma unroll
  for (int m = 0; m < 6; ++m) acc[m] = (v8f){0, 0, 0, 0, 0, 0, 0, 0};

  // This wave's K partition: 512 k = 32 chunks of 16 k.
  const int k0 = wave * (KDIM / 4);

  v2f a0[4], b0[4], a1[4], b1[4];
  load_chunk(Ap, Bp, k0, koff, a0, b0);       // chunk 0

  // 15 iterations x 2 chunks: compute chunks 0..29, prefetching 1..30
  for (int c = 0; c < 15; ++c) {
    const int kb = k0 + c * 32;
    load_chunk(Ap, Bp, kb + 16, koff, a1, b1);     // prefetch chunk 2c+1
    compute_chunk(a0, b0, sa, sb, acc);            // compute  chunk 2c
    load_chunk(Ap, Bp, kb + 32, koff, a0, b0);     // prefetch chunk 2c+2
    compute_chunk(a1, b1, sa, sb, acc);            // compute  chunk 2c+1
  }
  // Epilogue: a0 holds chunk 30; fetch chunk 31, drain both.
  load_chunk(Ap, Bp, k0 + 31 * 16, koff, a1, b1);
  compute_chunk(a0, b0, sa, sb, acc);
  compute_chunk(a1, b1, sa, sb, acc);

  // Cross-wave (K-partition) reduction through LDS: 4*6*8*32*4B = 24 KB
  __shared__ float red[4][6][8][32];
#pragma unroll
  for (int m = 0; m < 6; ++m)
#pragma unroll
    for (int r = 0; r < 8; ++r) red[wave][m][r][lane] = acc[m][r];
  __syncthreads();

  if (wave == 0) {
    const int n = lane & 15;
    const int mbase = (lane >> 4) << 3;       // C/D layout: lanes 16-31 hold M=8+r
#pragma unroll
    for (int r = 0; r < 8; ++r) {
      float g[6];
#pragma unroll
      for (int m = 0; m < 6; ++m)
        g[m] = red[0][m][r][lane] + red[1][m][r][lane] +
               red[2][m][r][lane] + red[3][m][r][lane];
      const float g11 = g[0], g41 = g[1], g32 = g[2];
      const float g62 = g[3], g44 = g[4], g23 = g[5];

      const float du = fmaxf(sqrtf(1.0f + 2.0f * g32 + g62), 1e-12f);
      const float ds = (g11 + g41) / (du * TEMP_F);
      const float dv = fmaxf(sqrtf(1.0f + 2.0f * g23 + g44), 1e-12f);
      const float dt = (g11 + g32) / (dv * TEMP_F);

      const int gi = bi * 16 + mbase + r;
      const int gj = bj * 16 + n;
      dself[gi * BDIM + gj] = ds;
      dother[gi * BDIM + gj] = dt;
    }
  }
}

// ---------------------------------------------------------------------------
// Kernel 3: final reduction.  Thread i owns row i:
//   vs_i = sum_j exp(dself) + 3*exp(dother);  sd_i = sum_j dself
//   loss = sum_i [ log(vs_i)/B - sd_i/B^2 ]
// ---------------------------------------------------------------------------
__global__ __launch_bounds__(128) void loss_kernel(
    const float* __restrict__ dself, const float* __restrict__ dother,
    float* __restrict__ out) {
  const int i = threadIdx.x;
  float se = 0.0f, sd = 0.0f;
  for (int j = 0; j < BDIM; ++j) {
    const float a = dself[i * BDIM + j];
    const float b = dother[i * BDIM + j];
    se += __expf(a) + 3.0f * __expf(b);
    sd += a;
  }
  const float contrib =
      __logf(se) * (1.0f / BDIM) - sd * (1.0f / (float)(BDIM * BDIM));
  __shared__ float red[128];
  red[i] = contrib;
  __syncthreads();
  for (int st = 64; st > 0; st >>= 1) {
    if (i < st) red[i] += red[i + st];
    __syncthreads();
  }
  if (i == 0) out[0] = red[0];
}

// ---------------------------------------------------------------------------
extern "C" void kernel_launch(void* const* d_in, const int* in_sizes, int n_in,
                              void* d_out, int out_size, void* d_ws,
                              size_t ws_size, hipStream_t stream) {
  (void)in_sizes; (void)n_in; (void)out_size; (void)ws_size;
  const float* feat = (const float*)d_in[0];
  float* out = (float*)d_out;

  char* ws = (char*)d_ws;
  float* invn = (float*)ws;                      // 128 floats
  float* dself = (float*)(ws + 1024);            // 128*128 floats
  float* dother = dself + BDIM * BDIM;           // 128*128 floats

  row_invnorm_kernel<<<BDIM, 256, 0, stream>>>(feat, invn);

  dim3 grid(BDIM / 16, BDIM / 16);               // 8x8 tiles
  gram_tile_kernel<<<grid, 128, 0, stream>>>(feat, invn, dself, dother);

  loss_kernel<<<1, 128, 0, stream>>>(dself, dother, out);
}